// Version3_MultiSub_Contrastive_29678224016216
// MI455X (gfx1250) — compile-verified
//
#include <hip/hip_runtime.h>
#include <hip/hip_bf16.h>
#include <math.h>

typedef float v2f __attribute__((ext_vector_type(2)));
typedef float v8f __attribute__((ext_vector_type(8)));

// Force global (addrspace 1) loads for weight streams: saddr + 32-bit voffset,
// tracked by LOADcnt only (no FLAT DScnt coupling with the LDS pipeline).
typedef const __attribute__((address_space(1))) float* gcfloat;
static __device__ __forceinline__ float gld(const float* p, unsigned i) {
  return ((gcfloat)p)[i];
}

#define BATCH    65536
#define HDIM     256
#define MID      300
#define MIDP     304           // 300 padded to 19 tiles of 16
#define THRESH   0.6f

#define MROWS    32            // rows per workgroup
#define NTHREADS 256
#define NWAVES   8

// LDS row strides (floats), padded so 16 row-parallel reads hit distinct banks
#define XSTR 260
#define HSTR 308
#define ESTR 260
#define CSTR 260
#define ASTR 516

static __device__ __forceinline__ v8f wmma_f32(v2f a, v2f b, v8f c) {
  // D = A(16x4 f32) * B(4x16 f32) + C(16x16 f32), full-precision f32 path
  return __builtin_amdgcn_wmma_f32_16x16x4_f32(false, a, false, b, (short)0, c,
                                               false, false);
}

// One encoder: E = relu(X @ W1 + b1) @ W2 + b2 for a [MROWS x 256] row tile.
static __device__ __forceinline__ void encoder(
    const float* __restrict__ Xg, const float* __restrict__ W1,
    const float* __restrict__ b1, const float* __restrict__ W2,
    const float* __restrict__ b2,
    float* __restrict__ sX, float* __restrict__ sH, float* __restrict__ E,
    int tid, int wave, int r16, int half, unsigned row0)
{
  // cooperative 128-bit load of the activation tile
  for (int t = tid; t < MROWS * (HDIM / 4); t += NTHREADS) {
    int r = t >> 6, c4 = t & 63;
    float4 v = ((const float4*)Xg)[(row0 + r) * (HDIM / 4) + c4];
    *(float4*)&sX[r * XSTR + c4 * 4] = v;
  }
  __syncthreads();

  // GEMM1: sH = relu(sX @ W1 + b1), [32 x 304]; cols >= 300 are zero via a
  // clamped always-valid index and a 0/1 mask multiply (no exec divergence).
  for (int nt = wave; nt < MIDP / 16; nt += NWAVES) {
    int      colN = nt * 16 + r16;
    bool     nv   = colN < MID;
    float    nvf  = nv ? 1.f : 0.f;
    unsigned colC = nv ? (unsigned)colN : (unsigned)(MID - 1);
    v8f acc0 = {}; v8f acc1 = {};
    for (int k0 = 0; k0 < HDIM; k0 += 4) {
      unsigned kk = (unsigned)(k0 + 2 * half);
      v2f a0 = *(const v2f*)&sX[r16 * XSTR + kk];
      v2f a1 = *(const v2f*)&sX[(16 + r16) * XSTR + kk];
      v2f b;
      b.x = gld(W1, kk * (unsigned)MID + colC) * nvf;
      b.y = gld(W1, (kk + 1u) * (unsigned)MID + colC) * nvf;
      acc0 = wmma_f32(a0, b, acc0);
      acc1 = wmma_f32(a1, b, acc1);
    }
    float bias = gld(b1, colC) * nvf;
    #pragma unroll
    for (int j = 0; j < 8; ++j) {
      int m = j + 8 * half;
      // relu(x) >= 0, so masking with *nvf equals the select; stays branch-free
      sH[m * HSTR + colN]        = fmaxf(acc0[j] + bias, 0.f) * nvf;
      sH[(16 + m) * HSTR + colN] = fmaxf(acc1[j] + bias, 0.f) * nvf;
    }
  }
  __syncthreads();

  // GEMM2: E = sH @ W2 + b2, [32 x 256], K = 300 (75 K-steps)
  for (int nt = wave; nt < HDIM / 16; nt += NWAVES) {
    unsigned colN = (unsigned)(nt * 16 + r16);
    v8f acc0 = {}; v8f acc1 = {};
    for (int k0 = 0; k0 < MID; k0 += 4) {
      unsigned kk = (unsigned)(k0 + 2 * half);
      v2f a0 = *(const v2f*)&sH[r16 * HSTR + kk];
      v2f a1 = *(const v2f*)&sH[(16 + r16) * HSTR + kk];
      v2f b;
      b.x = gld(W2, kk * (unsigned)HDIM + colN);
      b.y = gld(W2, (kk + 1u) * (unsigned)HDIM + colN);
      acc0 = wmma_f32(a0, b, acc0);
      acc1 = wmma_f32(a1, b, acc1);
    }
    float bias = gld(b2, colN);
    #pragma unroll
    for (int j = 0; j < 8; ++j) {
      int m = j + 8 * half;
      E[m * ESTR + colN]        = acc0[j] + bias;
      E[(16 + m) * ESTR + colN] = acc1[j] + bias;
    }
  }
  __syncthreads();
}

__global__ __launch_bounds__(NTHREADS)
void fused_cfe_kernel(
    const float* __restrict__ brics, const float* __restrict__ fgrp,
    const float* __restrict__ phar,
    const float* __restrict__ W1b, const float* __restrict__ b1b,
    const float* __restrict__ W2b, const float* __restrict__ b2b,
    const float* __restrict__ W1f, const float* __restrict__ b1f,
    const float* __restrict__ W2f, const float* __restrict__ b2f,
    const float* __restrict__ W1p, const float* __restrict__ b1p,
    const float* __restrict__ W2p, const float* __restrict__ b2p,
    const float* __restrict__ Wg,  const float* __restrict__ bg,
    const float* __restrict__ We,  const float* __restrict__ be,
    const float* __restrict__ Wf,  const float* __restrict__ bf,
    float* __restrict__ out)
{
  __shared__ float sX[MROWS * XSTR];        //  33,280 B  activation tile
  __shared__ float sH[MROWS * HSTR];        //  39,424 B  encoder mid (304 cols)
  __shared__ float sE[3 * MROWS * ESTR];    //  99,840 B  eb/ef/ep tiles
  __shared__ float sC[MROWS * CSTR];        //  33,280 B  common features
  __shared__ float sA2[MROWS * ASTR];       //  66,048 B  [weighted | common*enh]

  const int tid  = threadIdx.x;
  const int wave = tid >> 5;
  const int lane = tid & 31;
  const int r16  = lane & 15;   // M (A) / N (B,C,D) index within 16-wide half
  const int half = lane >> 4;   // selects K-pair (A,B) / M+8 (C,D)
  const unsigned row0 = blockIdx.x * MROWS;

  // ===== Phase 1: three encoders, fully unrolled so every weight pointer is
  // a direct kernel arg (keeps address-space inference => global_load saddr).
  encoder(brics, W1b, b1b, W2b, b2b, sX, sH, &sE[0],
          tid, wave, r16, half, row0);
  encoder(fgrp,  W1f, b1f, W2f, b2f, sX, sH, &sE[MROWS * ESTR],
          tid, wave, r16, half, row0);
  encoder(phar,  W1p, b1p, W2p, b2p, sX, sH, &sE[2 * MROWS * ESTR],
          tid, wave, r16, half, row0);

  // ====== Phase 2: per-row norms, sims, masked softmax, gates ======
  const float* Eb = &sE[0];
  const float* Ef = &sE[MROWS * ESTR];
  const float* Ep = &sE[2 * MROWS * ESTR];
  for (int i = 0; i < MROWS / NWAVES; ++i) {
    int r = wave * (MROWS / NWAVES) + i;
    float xb[8], xf[8], xp[8];
    float ssb = 0.f, ssf = 0.f, ssp = 0.f, d01 = 0.f, d02 = 0.f, d12 = 0.f;
    #pragma unroll
    for (int q = 0; q < 8; ++q) {
      int h = lane + 32 * q;
      float b_ = Eb[r * ESTR + h], f_ = Ef[r * ESTR + h], p_ = Ep[r * ESTR + h];
      xb[q] = b_; xf[q] = f_; xp[q] = p_;
      ssb += b_ * b_; ssf += f_ * f_; ssp += p_ * p_;
      d01 += b_ * f_; d02 += b_ * p_; d12 += f_ * p_;
    }
    // weight_generator logits: allc = [eb|ef|ep] (len 768) against Wg[768,3]
    float g0 = 0.f, g1 = 0.f, g2 = 0.f;
    #pragma unroll
    for (int q = 0; q < 8; ++q) {
      unsigned h = (unsigned)(lane + 32 * q);
      g0 += xb[q]*gld(Wg, h*3u+0u) + xf[q]*gld(Wg, (HDIM+h)*3u+0u) + xp[q]*gld(Wg, (2u*HDIM+h)*3u+0u);
      g1 += xb[q]*gld(Wg, h*3u+1u) + xf[q]*gld(Wg, (HDIM+h)*3u+1u) + xp[q]*gld(Wg, (2u*HDIM+h)*3u+1u);
      g2 += xb[q]*gld(Wg, h*3u+2u) + xf[q]*gld(Wg, (HDIM+h)*3u+2u) + xp[q]*gld(Wg, (2u*HDIM+h)*3u+2u);
    }
    #pragma unroll
    for (int off = 16; off > 0; off >>= 1) {
      ssb += __shfl_xor(ssb, off, 32);
      ssf += __shfl_xor(ssf, off, 32);
      ssp += __shfl_xor(ssp, off, 32);
      d01 += __shfl_xor(d01, off, 32);
      d02 += __shfl_xor(d02, off, 32);
      d12 += __shfl_xor(d12, off, 32);
      g0  += __shfl_xor(g0,  off, 32);
      g1  += __shfl_xor(g1,  off, 32);
      g2  += __shfl_xor(g2,  off, 32);
    }
    float nb  = fmaxf(sqrtf(ssb), 1e-12f);
    float nf  = fmaxf(sqrtf(ssf), 1e-12f);
    float np_ = fmaxf(sqrtf(ssp), 1e-12f);
    float s01 = d01 / (nb * nf), s02 = d02 / (nb * np_), s12 = d12 / (nf * np_);
    bool m01 = s01 > THRESH, m02 = s02 > THRESH, m12 = s12 > THRESH;
    bool has = m01 || m02 || m12;
    float mm = -3.4e38f;
    if (m01) mm = s01;
    if (m02) mm = fmaxf(mm, s02);
    if (m12) mm = fmaxf(mm, s12);
    if (!has) mm = 0.f;
    float e01 = m01 ? expf(s01 - mm) : 0.f;
    float e02 = m02 ? expf(s02 - mm) : 0.f;
    float e12 = m12 ? expf(s12 - mm) : 0.f;
    float esum = fmaxf(e01 + e02 + e12, 1e-12f);
    float w01 = e01 / esum, w02 = e02 / esum, w12 = e12 / esum;

    g0 += gld(bg, 0u); g1 += gld(bg, 1u); g2 += gld(bg, 2u);
    float gm  = fmaxf(g0, fmaxf(g1, g2));
    float eg0 = expf(g0 - gm), eg1 = expf(g1 - gm), eg2 = expf(g2 - gm);
    float ges = eg0 + eg1 + eg2;
    float fw0 = eg0 / ges, fw1 = eg1 / ges, fw2 = eg2 / ges;

    float ib = 1.f / nb, ifv = 1.f / nf, ip = 1.f / np_;
    #pragma unroll
    for (int q = 0; q < 8; ++q) {
      int h = lane + 32 * q;
      float b_ = xb[q], f_ = xf[q], p_ = xp[q];
      float c;
      if (has) {
        c  = ((b_ * ib) * (f_ * ifv) > THRESH) ? w01 * 0.5f * (b_ + f_) : 0.f;
        c += ((b_ * ib) * (p_ * ip ) > THRESH) ? w02 * 0.5f * (b_ + p_) : 0.f;
        c += ((f_ * ifv) * (p_ * ip) > THRESH) ? w12 * 0.5f * (f_ + p_) : 0.f;
      } else {
        c = (b_ + f_ + p_) * (1.f / 3.f);
      }
      sC[r * CSTR + h]  = c;                                 // common features
      sA2[r * ASTR + h] = b_ * fw0 + f_ * fw1 + p_ * fw2;    // gated sum
    }
  }
  __syncthreads();

  // ====== Phase 3: enh = sigmoid(C @ We + be); sA2[:,256:] = C * enh ======
  for (int nt = wave; nt < HDIM / 16; nt += NWAVES) {
    unsigned colN = (unsigned)(nt * 16 + r16);
    v8f acc0 = {}; v8f acc1 = {};
    for (int k0 = 0; k0 < HDIM; k0 += 4) {
      unsigned kk = (unsigned)(k0 + 2 * half);
      v2f a0 = *(const v2f*)&sC[r16 * CSTR + kk];
      v2f a1 = *(const v2f*)&sC[(16 + r16) * CSTR + kk];
      v2f b;
      b.x = gld(We, kk * (unsigned)HDIM + colN);
      b.y = gld(We, (kk + 1u) * (unsigned)HDIM + colN);
      acc0 = wmma_f32(a0, b, acc0);
      acc1 = wmma_f32(a1, b, acc1);
    }
    float bias = gld(be, colN);
    #pragma unroll
    for (int j = 0; j < 8; ++j) {
      int m = j + 8 * half;
      float eh0 = 1.f / (1.f + expf(-(acc0[j] + bias)));
      float eh1 = 1.f / (1.f + expf(-(acc1[j] + bias)));
      sA2[m * ASTR + HDIM + colN]        = sC[m * CSTR + colN] * eh0;
      sA2[(16 + m) * ASTR + HDIM + colN] = sC[(16 + m) * CSTR + colN] * eh1;
    }
  }
  __syncthreads();

  // ====== Phase 4: out = [weighted | C*enh] @ Wf + bf  (K = 512) ======
  for (int nt = wave; nt < HDIM / 16; nt += NWAVES) {
    unsigned colN = (unsigned)(nt * 16 + r16);
    v8f acc0 = {}; v8f acc1 = {};
    for (int k0 = 0; k0 < 2 * HDIM; k0 += 4) {
      unsigned kk = (unsigned)(k0 + 2 * half);
      v2f a0 = *(const v2f*)&sA2[r16 * ASTR + kk];
      v2f a1 = *(const v2f*)&sA2[(16 + r16) * ASTR + kk];
      v2f b;
      b.x = gld(Wf, kk * (unsigned)HDIM + colN);
      b.y = gld(Wf, (kk + 1u) * (unsigned)HDIM + colN);
      acc0 = wmma_f32(a0, b, acc0);
      acc1 = wmma_f32(a1, b, acc1);
    }
    float bias = gld(bf, colN);
    #pragma unroll
    for (int j = 0; j < 8; ++j) {
      unsigned m = (unsigned)(j + 8 * half);
      out[(row0 + m) * (unsigned)HDIM + colN]       = acc0[j] + bias;
      out[(row0 + 16u + m) * (unsigned)HDIM + colN] = acc1[j] + bias;
    }
  }
}

extern "C" void kernel_launch(void* const* d_in, const int* in_sizes, int n_in,
                              void* d_out, int out_size, void* d_ws, size_t ws_size,
                              hipStream_t stream) {
  (void)in_sizes; (void)n_in; (void)out_size; (void)d_ws; (void)ws_size;
  const float* brics = (const float*)d_in[0];
  const float* fgrp  = (const float*)d_in[1];
  const float* phar  = (const float*)d_in[2];
  const float* W1b = (const float*)d_in[3];
  const float* b1b = (const float*)d_in[4];
  const float* W2b = (const float*)d_in[5];
  const float* b2b = (const float*)d_in[6];
  const float* W1f = (const float*)d_in[7];
  const float* b1f = (const float*)d_in[8];
  const float* W2f = (const float*)d_in[9];
  const float* b2f = (const float*)d_in[10];
  const float* W1p = (const float*)d_in[11];
  const float* b1p = (const float*)d_in[12];
  const float* W2p = (const float*)d_in[13];
  const float* b2p = (const float*)d_in[14];
  const float* Wg  = (const float*)d_in[15];
  const float* bg  = (const float*)d_in[16];
  const float* We  = (const float*)d_in[17];
  const float* be  = (const float*)d_in[18];
  const float* Wf  = (const float*)d_in[19];
  const float* bf  = (const float*)d_in[20];
  float* out = (float*)d_out;

  dim3 grid(BATCH / MROWS);   // 2048 workgroups
  dim3 block(NTHREADS);       // 8 wave32s
  fused_cfe_kernel<<<grid, block, 0, stream>>>(
      brics, fgrp, phar,
      W1b, b1b, W2b, b2b, W1f, b1f, W2f, b2f, W1p, b1p, W2p, b2p,
      Wg, bg, We, be, Wf, bf, out);
}